// HungarianMatcher_77472620085888
// MI455X (gfx1250) — compile-verified
//
#include <hip/hip_runtime.h>
#include <math.h>

// Problem constants (match the JAX reference).
#define Bc 128
#define Qc 900
#define Nc 300
#define Cc 92
#define KCH 23   // Cc / 4 K-chunks for V_WMMA_F32_16X16X4_F32

typedef float v2f __attribute__((ext_vector_type(2)));
typedef float v8f __attribute__((ext_vector_type(8)));

// One wave32 per 16-query strip of one batch image.
// A = exp(logits - rowmax) [16 x 92], B = one-hot(labels) [92 x 16] built in
// VGPRs -> D accumulates the *exact* softmax-numerator gather via f32 WMMA.
__global__ __launch_bounds__(32)
void hungarian_cost_kernel(const float* __restrict__ logits,   // [B,Q,C]
                           const float* __restrict__ pboxes,   // [B,Q,4] cxcywh
                           const float* __restrict__ tboxes,   // [B,N,4] xyxy
                           const int*   __restrict__ tlabels,  // [B,N]
                           float* __restrict__ out)            // [B,Q,N]
{
  const int  lane = threadIdx.x;      // 0..31
  const int  nlo  = lane & 15;
  const bool hiH  = lane >= 16;
  const int  b    = blockIdx.y;
  const int  q0   = blockIdx.x * 16;

  // ---------- A-matrix fragments: exp(logit - max) for query row q0+nlo ----
  // A 16x4 f32 layout: lane L -> M = L&15; VGPR0/1 hold K = (L<16?0:2)+{0,1}.
  int qa = q0 + nlo; if (qa >= Qc) qa = Qc - 1;          // clamp (store-guarded)
  const float4* lrow = (const float4*)(logits + ((size_t)b * Qc + qa) * Cc);

  float mx = -3.402823466e38f;
#pragma unroll
  for (int k = 0; k < KCH; ++k) {
    float4 v = lrow[k];
    mx = fmaxf(mx, fmaxf(fmaxf(v.x, v.y), fmaxf(v.z, v.w)));
  }
  float sum = 0.0f;
  v2f afrag[KCH];
#pragma unroll
  for (int k = 0; k < KCH; ++k) {
    float4 v = lrow[k];
    float e0 = __expf(v.x - mx);
    float e1 = __expf(v.y - mx);
    float e2 = __expf(v.z - mx);
    float e3 = __expf(v.w - mx);
    sum += (e0 + e1) + (e2 + e3);
    afrag[k].x = hiH ? e2 : e0;
    afrag[k].y = hiH ? e3 : e1;
  }
  float rinv = 1.0f / sum;   // softmax denominator reciprocal for row q0+nlo

  // C/D 16x16 f32 layout: lane L, VGPR r -> (m = r + (L<16?0:8), n = L&15).
  const int mbase = hiH ? 8 : 0;
  float rinvm[8];
#pragma unroll
  for (int r = 0; r < 8; ++r)
    rinvm[r] = __shfl(rinv, mbase + r, 32);   // lane m holds rinv of row m

  // ---------- pred boxes for this lane's 8 query rows (reused over all n) --
  float pcx[8], pcy[8], pw[8], ph[8];
#pragma unroll
  for (int r = 0; r < 8; ++r) {
    int q = q0 + mbase + r; if (q >= Qc) q = Qc - 1;
    float4 pb = ((const float4*)pboxes)[(size_t)b * Qc + q];
    pcx[r] = pb.x; pcy[r] = pb.y; pw[r] = pb.z; ph[r] = pb.w;
  }

  const int koff = hiH ? 2 : 0;   // B 4x16: lane half selects K rows {0,1}/{2,3}

  for (int nt = 0; nt < (Nc + 15) / 16; ++nt) {
    const int  nn     = nt * 16 + nlo;
    const bool nvalid = nn < Nc;
    const int  ncl    = nvalid ? nn : Nc - 1;

    float4 tb  = ((const float4*)tboxes)[(size_t)b * Nc + ncl];   // xyxy
    int    lbl = tlabels[(size_t)b * Nc + ncl];
    float tcx = 0.5f * (tb.x + tb.z), tcy = 0.5f * (tb.y + tb.w);
    float tw  = tb.z - tb.x,          th2 = tb.w - tb.y;
    float tarea = tw * th2;

    // ---- exact one-hot gather via f32 WMMA: D[m,n] = exp(logit[m,lbl[n]]-mx)
    v8f d = {0.f, 0.f, 0.f, 0.f, 0.f, 0.f, 0.f, 0.f};
#pragma unroll
    for (int k = 0; k < KCH; ++k) {
      const int cb = 4 * k + koff;
      v2f bf;
      bf.x = (lbl == cb)     ? 1.0f : 0.0f;
      bf.y = (lbl == cb + 1) ? 1.0f : 0.0f;
      d = __builtin_amdgcn_wmma_f32_16x16x4_f32(false, afrag[k], false, bf,
                                                (short)0, d, false, false);
    }

    // ---- combine class + L1 + GIoU, stream out with NT stores -------------
#pragma unroll
    for (int r = 0; r < 8; ++r) {
      const int q = q0 + mbase + r;
      float clsc = -d[r] * rinvm[r];                         // -prob

      float l1 = fabsf(pcx[r] - tcx) + fabsf(pcy[r] - tcy)
               + fabsf(pw[r]  - tw ) + fabsf(ph[r]  - th2);

      float hw = 0.5f * pw[r], hh = 0.5f * ph[r];
      float x0 = pcx[r] - hw, y0 = pcy[r] - hh;
      float x1 = pcx[r] + hw, y1 = pcy[r] + hh;

      float iw = fminf(x1, tb.z) - fmaxf(x0, tb.x);
      float ih = fminf(y1, tb.w) - fmaxf(y0, tb.y);
      iw = fmaxf(iw, 0.0f); ih = fmaxf(ih, 0.0f);
      float inter = iw * ih;
      float uni   = pw[r] * ph[r] + tarea - inter;
      float iou   = inter / uni;

      float ew = fmaxf(x1, tb.z) - fminf(x0, tb.x);
      float eh = fmaxf(y1, tb.w) - fminf(y0, tb.y);
      ew = fmaxf(ew, 0.0f); eh = fmaxf(eh, 0.0f);
      float ea   = ew * eh;
      float giou = iou - (ea - uni) / ea;

      float cost = 5.0f * l1 + clsc - 2.0f * giou;
      if (nvalid && q < Qc)
        __builtin_nontemporal_store(cost,
            out + ((size_t)b * Qc + q) * Nc + nn);
    }
  }
}

extern "C" void kernel_launch(void* const* d_in, const int* in_sizes, int n_in,
                              void* d_out, int out_size, void* d_ws, size_t ws_size,
                              hipStream_t stream) {
  const float* logits  = (const float*)d_in[0];  // pred_logits [B,Q,C] f32
  const float* pboxes  = (const float*)d_in[1];  // pred_boxes  [B,Q,4] f32
  const float* tboxes  = (const float*)d_in[2];  // tgt_boxes   [B,N,4] f32
  const int*   tlabels = (const int*)  d_in[3];  // tgt_labels  [B,N]   i32
  float*       out     = (float*)d_out;          // [B,Q,N] f32

  dim3 grid((Qc + 15) / 16, Bc);   // 57 x 128 single-wave workgroups
  hungarian_cost_kernel<<<grid, 32, 0, stream>>>(logits, pboxes, tboxes,
                                                 tlabels, out);
  (void)in_sizes; (void)n_in; (void)out_size; (void)d_ws; (void)ws_size;
}